// QDFRCell_5411658793408
// MI455X (gfx1250) — compile-verified
//
#include <hip/hip_runtime.h>
#include <stdint.h>

// ---------------------------------------------------------------------------
// QDFRCell forward for MI455X (gfx1250).
//
// Roofline: 128 MB read (prev) + 128 MB write (out) @ 23.3 TB/s => ~11 us
// floor; arithmetic is ~0.7 GFLOP => deeply memory-bound. Strategy:
//   * rank-1 "matmul" (x[B,1] @ q_mask[1,N]) done on the matrix pipe with
//     V_WMMA_F32_16X16X4_F32, one WMMA per 16x16 output tile (K=1 padded
//     with zeros into K=4),
//   * q_mask (8 KB, reused by every tile in a block) staged into LDS with
//     CDNA5 async DMA (global_load_async_to_lds_b128 + s_wait_asynccnt),
//   * prev/out streamed once with NT temporal hints; per C-tile VGPR layout
//     each b32 access is two fully-used 64B segments per wave, addressed as
//     one base + immediate offsets r*N*4 (fits 24-bit IOFFSET).
// ---------------------------------------------------------------------------

typedef float v2f __attribute__((ext_vector_type(2)));
typedef float v8f __attribute__((ext_vector_type(8)));

#define N_MAX 2048

// fake_quant(v, 0, 2) with 8-bit levels: scale = 2/255.
__device__ __forceinline__ float fq2(float v) {
    const float scale = 2.0f / 255.0f;
    const float inv   = 255.0f / 2.0f;   // 127.5
    float xc = fminf(fmaxf(v, 0.0f), 2.0f);
    return rintf(xc * inv) * scale;      // rintf == round-half-even (jnp.round)
}

// ---------------------------------------------------------------------------
// Main kernel (first in file so the disasm snippet shows it).
// Block = 256 threads = 8 waves; block owns a 16-row band x all N columns.
// Wave w handles 16x16 column-tiles w, w+8, ... of the band.
// ---------------------------------------------------------------------------
__global__ void __launch_bounds__(256)
qdfr_wmma_kernel(const float* __restrict__ x,
                 const float* __restrict__ prev,
                 const float* __restrict__ qmask,
                 float* __restrict__ out,
                 int n) {
    __shared__ __align__(16) float smask[N_MAX];
    const int tid  = threadIdx.x;
    const int lane = tid & 31;          // wave32
    const int wv   = tid >> 5;          // wave id 0..7
    const int nchunks = n >> 2;

    // ---- Stage q_mask into LDS via CDNA5 async DMA (ASYNCcnt path). ----
    {
        const uint64_t gbase = (uint64_t)(uintptr_t)qmask;
        for (int c = tid; c < nchunks; c += 256) {
            uint32_t lds_off = (uint32_t)(uintptr_t)(&smask[c << 2]);
            uint32_t goff    = (uint32_t)(c << 4);
            asm volatile("global_load_async_to_lds_b128 %0, %1, %2"
                         :
                         : "v"(lds_off), "v"(goff), "s"(gbase)
                         : "memory");
        }
        asm volatile("s_wait_asynccnt 0" ::: "memory");
        __syncthreads();
    }

    // ---- A operand: xq for the block's 16 rows, in column K=0. ----
    // A(16x4) layout: VGPR0 lanes 0-15 = K=0 rows M=0..15; everything else 0.
    const float scale_in = 0.497f / 255.0f;
    const float inv_in   = 255.0f / 0.497f;
    const int   row0     = blockIdx.x << 4;

    float xv = x[row0 + (lane & 15)];
    float xq = rintf(fminf(fmaxf(xv, 0.0f), 0.497f) * inv_in) * scale_in;
    v2f a;
    a.x = (lane < 16) ? xq : 0.0f;      // K=0 (lanes 0-15) / K=2 (zeroed)
    a.y = 0.0f;                          // K=1 / K=3 zeroed

    const int half = lane >> 4;          // C/D rows M = r + 8*half
    const int colL = lane & 15;          // C/D column N = lane % 16
    const int ntiles = n >> 4;

    for (int t = wv; t < ntiles; t += 8) {
        const int col0 = t << 4;

        // B(4x16): row K=0 = q_mask[col0..col0+15] in lanes 0-15; rest 0.
        v2f b;
        b.x = (lane < 16) ? smask[col0 + colL] : 0.0f;   // ds_load_b32
        b.y = 0.0f;
        v8f c = {};
        // D[M][N] = xq[row0+M] * q_mask[col0+N]  (rank-1 via matrix pipe)
        v8f d = __builtin_amdgcn_wmma_f32_16x16x4_f32(
            /*neg_a=*/false, a, /*neg_b=*/false, b,
            /*c_mod=*/(short)0, c, /*reuse_a=*/false, /*reuse_b=*/false);

        // Lane's 8 elements: (row0 + r + 8*half, col0 + colL), r = 0..7.
        const float* pbase = prev + (size_t)(row0 + 8 * half) * n + col0 + colL;
        float*       obase = out  + (size_t)(row0 + 8 * half) * n + col0 + colL;

        #pragma unroll
        for (int r = 0; r < 8; ++r) {
            float pb = __builtin_nontemporal_load(pbase + (size_t)r * n);
            float qv = fq2(d[r]);            // fake_quant(vec_x, 0, 2)
            float bq = fq2(0.2f * pb);       // fake_quant(0.2*prev, 0, 2)
            float oo = fmaxf(qv + pb - bq, 0.0f);          // relu
            __builtin_nontemporal_store(fq2(oo), obase + (size_t)r * n);
        }
    }
}

// ---------------------------------------------------------------------------
// Kernel 1: dynamic fake-quant of mask[1,N] into workspace.
// ---------------------------------------------------------------------------
__global__ void __launch_bounds__(256)
qmask_kernel(const float* __restrict__ mask, float* __restrict__ qmask, int n) {
    __shared__ float smin[256];
    __shared__ float smax[256];
    const int tid = threadIdx.x;

    float mn =  3.402823466e38f;
    float mx = -3.402823466e38f;
    for (int j = tid; j < n; j += 256) {
        float v = mask[j];
        mn = fminf(mn, v);
        mx = fmaxf(mx, v);
    }
    smin[tid] = mn; smax[tid] = mx;
    __syncthreads();
    for (int s = 128; s > 0; s >>= 1) {
        if (tid < s) {
            smin[tid] = fminf(smin[tid], smin[tid + s]);
            smax[tid] = fmaxf(smax[tid], smax[tid + s]);
        }
        __syncthreads();
    }
    const float gmn = smin[0];
    const float gmx = smax[0];
    const float scale = (gmx - gmn) * (1.0f / 255.0f);
    const float inv   = 255.0f / (gmx - gmn);
    for (int j = tid; j < n; j += 256) {
        float v = fminf(fmaxf(mask[j], gmn), gmx);
        qmask[j] = rintf((v - gmn) * inv) * scale + gmn;
    }
}

// ---------------------------------------------------------------------------
// Launch
// ---------------------------------------------------------------------------
extern "C" void kernel_launch(void* const* d_in, const int* in_sizes, int n_in,
                              void* d_out, int out_size, void* d_ws, size_t ws_size,
                              hipStream_t stream) {
    const float* x    = (const float*)d_in[0];   // [B,1]
    const float* prev = (const float*)d_in[1];   // [B,N]
    const float* mask = (const float*)d_in[2];   // [1,N]
    float*       out  = (float*)d_out;           // [B,N]
    float*       qm   = (float*)d_ws;            // N floats of scratch

    const int B = in_sizes[0];
    const int N = in_sizes[2];

    qmask_kernel<<<1, 256, 0, stream>>>(mask, qm, N);

    // One block per 16-row band (B = 16384 -> 1024 blocks of 8 waves).
    qdfr_wmma_kernel<<<B >> 4, 256, 0, stream>>>(x, prev, qm, out, N);
}